// EquidistantDiscreteContinuousConv2d_89507118449173
// MI455X (gfx1250) — compile-verified
//
#include <hip/hip_runtime.h>
#include <hip/hip_bf16.h>

// DISCO depthwise conv, MI455X (gfx1250).
// Stage 1 (1 wave): assemble 16x25 effective kernel via V_WMMA_F32_16X16X4_F32.
// Stage 2: HBM-bound streaming depthwise 5x5 stride-2 conv.
//   4 outputs/thread, aligned b128 row-window loads, nontemporal b128 stores.
//   ~320 MB HBM traffic -> ~14 us roofline; instruction issue tuned below roof.

typedef __attribute__((ext_vector_type(2)))  float  v2f;
typedef __attribute__((ext_vector_type(4)))  float  v4f;
typedef __attribute__((ext_vector_type(8)))  float  v8f;

#define QW 3.814697265625e-06f   // DOMAIN area / (H_IN*W_IN) = 4/2^20

// ---------------------------------------------------------------------------
// Stage 1: keff[16][32] = QW * (weight(16x3) x psi(3x25)), taps 25..31 = 0.
// One wave32 block. Two WMMA tiles cover tap columns 0..15 and 16..24.
// ---------------------------------------------------------------------------
__global__ __launch_bounds__(32)
void disco_assemble(const float* __restrict__ w,     // (16,1,3)
                    const float* __restrict__ psi,   // (3,5,5) flattened (3,25)
                    float* __restrict__ keff)        // (16,32)
{
  const int lane = threadIdx.x;      // 0..31
  const int half = lane >> 4;        // 0: lanes 0-15, 1: lanes 16-31
  const int l15  = lane & 15;

  for (int tile = 0; tile < 2; ++tile) {
    const int  col = l15 + (tile << 4);   // tap column 0..31
    const bool ok  = col < 25;
    v8f acc = {};

    // 32-bit A 16x4 layout: lanes 0-15 hold K0(v0),K1(v1); lanes 16-31 hold K2,K3.
    v2f A, B;
    A.x = (half == 0) ? w[l15 * 3 + 0] : w[l15 * 3 + 2];
    A.y = (half == 0) ? w[l15 * 3 + 1] : 0.0f;
    // B 4x16: lanes 0-15 rows K0(v0),K1(v1); lanes 16-31 rows K2(v0),K3(v1); N = l15.
    B.x = ok ? ((half == 0) ? psi[0 * 25 + col] : psi[2 * 25 + col]) : 0.0f;
    B.y = (ok && half == 0) ? psi[1 * 25 + col] : 0.0f;
    acc = __builtin_amdgcn_wmma_f32_16x16x4_f32(false, A, false, B,
                                                (short)0, acc, false, false);

    // C/D 16x16 f32 layout: VGPR r -> (M=r, N=lane) lanes 0-15, (M=r+8) lanes 16-31.
#pragma unroll
    for (int r = 0; r < 8; ++r) {
      const int row = r + (half << 3);       // output channel
      keff[row * 32 + col] = QW * acc[r];    // zero for padded taps (B cols were 0)
    }
  }
}

// ---------------------------------------------------------------------------
// Stage 2: depthwise 5x5 stride-2 conv, zero padding 2.
// Block = 128 threads = one output row of one (b,c) plane -> c block-uniform,
// taps/bias lower to SMEM scalar loads. Each thread: 4 outputs (j0..j0+3),
// per input row one aligned 16-col window = 4x global_load_b128.
// Grid = 4*16*512 = 32768 blocks.
// ---------------------------------------------------------------------------
__global__ __launch_bounds__(128)
void disco_conv(const float* __restrict__ x,      // (4,16,1024,1024)
                const float* __restrict__ keff,   // (16,32)
                const float* __restrict__ bias,   // (16)
                float* __restrict__ out)          // (4,16,512,512)
{
  const int bid = blockIdx.x;
  const int i = bid & 511;
  const int c = (bid >> 9) & 15;
  const int b = bid >> 13;
  const int tid = (int)threadIdx.x;    // 0..127
  const int j0 = tid << 2;             // first of 4 output columns

  float kk[25];
#pragma unroll
  for (int t = 0; t < 25; ++t) kk[t] = keff[(c << 5) + t];   // uniform -> SGPR
  const float bv = bias[c];

  const float* plane = x + ((size_t)((b << 4) + c) << 20);   // 1024*1024 plane
  const int i2 = 2 * i - 2;                                  // top input row

  float a0 = bv, a1 = bv, a2 = bv, a3 = bv;

  if (i >= 1 && i <= 510 && tid >= 1 && tid <= 126) {
    // Interior fast path.
    // Window cols s..s+15, s = 2*j0-4 (16B aligned). Output q tap dw reads
    // window index 2 + 2q + dw (max 12 <= 15).
    const int s = (j0 << 1) - 4;
    const float* rowp = plane + i2 * 1024 + s;
    __builtin_prefetch(rowp + 6 * 1024, 0, 3);   // speculative global_prefetch_b8
#pragma unroll
    for (int dh = 0; dh < 5; ++dh) {
      const v4f* rp = (const v4f*)rowp;
      const v4f w0 = rp[0], w1 = rp[1], w2 = rp[2], w3 = rp[3];
      const float win[16] = {w0.x, w0.y, w0.z, w0.w,
                             w1.x, w1.y, w1.z, w1.w,
                             w2.x, w2.y, w2.z, w2.w,
                             w3.x, w3.y, w3.z, w3.w};
#pragma unroll
      for (int dw = 0; dw < 5; ++dw) {
        const float kv = kk[dh * 5 + dw];
        a0 = fmaf(kv, win[2 + dw], a0);
        a1 = fmaf(kv, win[4 + dw], a1);
        a2 = fmaf(kv, win[6 + dw], a2);
        a3 = fmaf(kv, win[8 + dw], a3);
      }
      rowp += 1024;
    }
  } else {
    // Border: bounds-checked scalar loads (zero padding).
    float aq[4] = {bv, bv, bv, bv};
#pragma unroll
    for (int q = 0; q < 4; ++q) {
      const int j2 = 2 * (j0 + q) - 2;
#pragma unroll
      for (int dh = 0; dh < 5; ++dh) {
        const int r = i2 + dh;
        if ((unsigned)r < 1024u) {
#pragma unroll
          for (int dw = 0; dw < 5; ++dw) {
            const int cc = j2 + dw;
            if ((unsigned)cc < 1024u)
              aq[q] = fmaf(kk[dh * 5 + dw], plane[r * 1024 + cc], aq[q]);
          }
        }
      }
    }
    a0 = aq[0]; a1 = aq[1]; a2 = aq[2]; a3 = aq[3];
  }

  v4f* op = (v4f*)(out + (((size_t)((b << 4) + c) << 18) + (i << 9) + j0));
  const v4f res = {a0, a1, a2, a3};
  __builtin_nontemporal_store(res, op);   // write-once output: NT global_store_b128
}

extern "C" void kernel_launch(void* const* d_in, const int* in_sizes, int n_in,
                              void* d_out, int out_size, void* d_ws, size_t ws_size,
                              hipStream_t stream) {
  const float* x      = (const float*)d_in[0];   // (4,16,1024,1024)
  const float* weight = (const float*)d_in[1];   // (16,1,3)
  const float* bias   = (const float*)d_in[2];   // (16,)
  const float* psi    = (const float*)d_in[3];   // (3,5,5)
  float*       outp   = (float*)d_out;           // (4,16,512,512)
  float*       keff   = (float*)d_ws;            // 16*32 floats scratch

  disco_assemble<<<1, 32, 0, stream>>>(weight, psi, keff);
  disco_conv<<<32768, 128, 0, stream>>>(x, keff, bias, outp);
}